// Correlation3D_1589137900207
// MI455X (gfx1250) — compile-verified
//
#include <hip/hip_runtime.h>

// CDNA5 / gfx1250. wave32. WMMA f32 16x16x4 used as an exact fp32
// 4-channel column-sum reducer (A = all ones).
typedef float v2f __attribute__((ext_vector_type(2)));
typedef float v8f __attribute__((ext_vector_type(8)));

#define VN 96        // volume edge
#define NC 32        // channels
#define BX 4         // tile x
#define BY 4         // tile y
#define BT 16        // tile t (innermost/contiguous)
#define NTHREADS 256 // 8 wave32
#define SCALE (1.0f / 27.0f)

// LDS layout (float offsets), arrays overlapped across phases:
//  S1 [4][10][10][20] = 8000  (t-pass of fix halo)     @ 0
//  BF [4][ 8][ 8][20] = 5120  (final fix box, overlays S1)
//  S2 [4][10][ 8][20] = 6400  (y-pass)                 @ 8000
//  M1 [4][ 6][ 6][16] = 2304  (t-pass of mov, overlays S2)
//  M2 [4][ 6][ 4][16] = 1536                           @ 10304
//  PM [4][ 4][ 4][16] = 1024                           @ 11840
#define S1_OFF 0
#define BF_OFF 0
#define S2_OFF 8000
#define M1_OFF 8000
#define M2_OFF 10304
#define PM_OFF 11840
#define SMEM_FLOATS 14400   // 57.6 KB

__device__ __forceinline__ float ldg_vol(const float* __restrict__ p,
                                         int c, int x, int y, int t) {
    if ((unsigned)x < (unsigned)VN && (unsigned)y < (unsigned)VN &&
        (unsigned)t < (unsigned)VN)
        return p[((c * VN + x) * VN + y) * VN + t];
    return 0.0f;  // zero padding, matches lax.conv padding
}

__global__ __launch_bounds__(NTHREADS)
void corr3d_wmma_kernel(const float* __restrict__ mov,
                        const float* __restrict__ fix,
                        float* __restrict__ out) {
    __shared__ float smem[SMEM_FLOATS];

    const int tid  = threadIdx.x;
    const int lane = tid & 31;
    const int w    = tid >> 5;       // wave id 0..7
    const int col  = lane & 15;      // WMMA column / voxel-in-16

    const int x0 = blockIdx.z * BX;
    const int y0 = blockIdx.y * BY;
    const int t0 = blockIdx.x * BT;

    // Each wave owns 32 voxels: vA = w*32+col (both lane halves share it),
    // vB = vA+16. Low lanes carry channels {0,1} of the group, high {2,3}.
    const int vA = w * 32 + col;
    const int vB = vA + 16;
    const int lxA = vA >> 6, lyA = (vA >> 4) & 3, ltA = vA & 15;
    const int lxB = vB >> 6, lyB = (vB >> 4) & 3, ltB = vB & 15;
    const int cc0 = (lane < 16) ? 0 : 2;
    const int cc1 = cc0 + 1;

    float accA[27], accB[27];
#pragma unroll
    for (int d = 0; d < 27; ++d) { accA[d] = 0.0f; accB[d] = 0.0f; }

    const v2f a_ones = {1.0f, 1.0f};  // all-ones A => D[m,n] = sum_k B[k,n]

    for (int cg = 0; cg < 8; ++cg) {          // 4 channels per group
        __syncthreads();                      // smem reuse fence
        // ---- fix box filter, separable. Phase 1: t-pass into S1 ----
        for (int idx = tid; idx < 8000; idx += NTHREADS) {
            int c4 = idx / 2000, r = idx - c4 * 2000;
            int ix = r / 200, rr = r - ix * 200;
            int iy = rr / 20, it = rr - iy * 20;
            int ch = cg * 4 + c4;
            int gx = x0 - 3 + ix, gy = y0 - 3 + iy, gt = t0 - 3 + it;
            float v = ldg_vol(fix, ch, gx, gy, gt)
                    + ldg_vol(fix, ch, gx, gy, gt + 1)
                    + ldg_vol(fix, ch, gx, gy, gt + 2);
            smem[S1_OFF + ((c4 * 10 + ix) * 10 + iy) * 20 + it] = v;
        }
        __syncthreads();
        // ---- Phase 2: y-pass into S2 ----
        for (int idx = tid; idx < 6400; idx += NTHREADS) {
            int c4 = idx / 1600, r = idx - c4 * 1600;
            int ix = r / 160, rr = r - ix * 160;
            int iy = rr / 20, it = rr - iy * 20;
            const float* s1 = &smem[S1_OFF + ((c4 * 10 + ix) * 10 + iy) * 20 + it];
            smem[S2_OFF + ((c4 * 10 + ix) * 8 + iy) * 20 + it] =
                s1[0] + s1[20] + s1[40];
        }
        __syncthreads();
        // ---- Phase 3: x-pass into BF (overlays S1, which is dead) ----
        for (int idx = tid; idx < 5120; idx += NTHREADS) {
            int c4 = idx / 1280, r = idx - c4 * 1280;
            int ix = r / 160, rr = r - ix * 160;
            int iy = rr / 20, it = rr - iy * 20;
            const float* s2 = &smem[S2_OFF + ((c4 * 10 + ix) * 8 + iy) * 20 + it];
            smem[BF_OFF + ((c4 * 8 + ix) * 8 + iy) * 20 + it] =
                s2[0] + s2[160] + s2[320];
        }
        __syncthreads();
        // ---- mov box filter. Phase 4: t-pass into M1 (overlays S2) ----
        for (int idx = tid; idx < 2304; idx += NTHREADS) {
            int c4 = idx / 576, r = idx - c4 * 576;
            int ix = r / 96, rr = r - ix * 96;
            int iy = rr / 16, lt = rr - iy * 16;
            int ch = cg * 4 + c4;
            int gx = x0 - 1 + ix, gy = y0 - 1 + iy, gt = t0 + lt - 1;
            float v = ldg_vol(mov, ch, gx, gy, gt)
                    + ldg_vol(mov, ch, gx, gy, gt + 1)
                    + ldg_vol(mov, ch, gx, gy, gt + 2);
            smem[M1_OFF + ((c4 * 6 + ix) * 6 + iy) * 16 + lt] = v;
        }
        __syncthreads();
        // ---- Phase 5: y-pass into M2 ----
        for (int idx = tid; idx < 1536; idx += NTHREADS) {
            int c4 = idx / 384, r = idx - c4 * 384;
            int ix = r / 64, rr = r - ix * 64;
            int ly = rr / 16, lt = rr - ly * 16;
            const float* m1 = &smem[M1_OFF + ((c4 * 6 + ix) * 6 + ly) * 16 + lt];
            smem[M2_OFF + ((c4 * 6 + ix) * 4 + ly) * 16 + lt] =
                m1[0] + m1[16] + m1[32];
        }
        __syncthreads();
        // ---- Phase 6: x-pass into PM ----
        for (int idx = tid; idx < 1024; idx += NTHREADS) {
            int c4 = idx / 256, r = idx - c4 * 256;
            int lx = r / 64, rr = r - lx * 64;
            int ly = rr / 16, lt = rr - ly * 16;
            const float* m2 = &smem[M2_OFF + ((c4 * 6 + lx) * 4 + ly) * 16 + lt];
            smem[PM_OFF + ((c4 * 4 + lx) * 4 + ly) * 16 + lt] =
                m2[0] + m2[64] + m2[128];
        }
        __syncthreads();

        // ---- Phase 7: 27-displacement correlation, channel reduction via
        // WMMA f32 16x16x4 with all-ones A (exact fp32 column sums). ----
        const float pmA0 = smem[PM_OFF + ((cc0 * 4 + lxA) * 4 + lyA) * 16 + ltA];
        const float pmA1 = smem[PM_OFF + ((cc1 * 4 + lxA) * 4 + lyA) * 16 + ltA];
        const float pmB0 = smem[PM_OFF + ((cc0 * 4 + lxB) * 4 + lyB) * 16 + ltB];
        const float pmB1 = smem[PM_OFF + ((cc1 * 4 + lxB) * 4 + lyB) * 16 + ltB];

#pragma unroll
        for (int i = 0; i < 3; ++i)
#pragma unroll
        for (int j = 0; j < 3; ++j)
#pragma unroll
        for (int k = 0; k < 3; ++k) {
            const int d = (i * 3 + j) * 3 + k;
            const int pA0 = ((cc0 * 8 + (lxA + 2*i)) * 8 + (lyA + 2*j)) * 20 + (ltA + 2*k);
            const int pA1 = ((cc1 * 8 + (lxA + 2*i)) * 8 + (lyA + 2*j)) * 20 + (ltA + 2*k);
            const int pB0 = ((cc0 * 8 + (lxB + 2*i)) * 8 + (lyB + 2*j)) * 20 + (ltB + 2*k);
            const int pB1 = ((cc1 * 8 + (lxB + 2*i)) * 8 + (lyB + 2*j)) * 20 + (ltB + 2*k);
            v2f bA = { pmA0 * smem[BF_OFF + pA0], pmA1 * smem[BF_OFF + pA1] };
            v2f bB = { pmB0 * smem[BF_OFF + pB0], pmB1 * smem[BF_OFF + pB1] };
            v8f z = {};
            // D = 1s * B + 0 : every row (and both lane halves) of D holds
            // the 4-channel sum for voxel column (lane & 15).
            v8f rA = __builtin_amdgcn_wmma_f32_16x16x4_f32(
                false, a_ones, false, bA, (short)0, z, false, false);
            v8f rB = __builtin_amdgcn_wmma_f32_16x16x4_f32(
                false, a_ones, false, bB, (short)0, z, false, false);
            accA[d] += rA[0];
            accB[d] += rB[0];
        }
    }

    // ---- store: lane owns voxel w*32+lane; low lanes from accA, high accB ----
    const int v  = w * 32 + lane;
    const int lx = v >> 6, ly = (v >> 4) & 3, lt = v & 15;
    const int x = x0 + lx, y = y0 + ly, t = t0 + lt;
#pragma unroll
    for (int d = 0; d < 27; ++d) {
        float val = ((lane < 16) ? accA[d] : accB[d]) * SCALE;
        out[((d * VN + x) * VN + y) * VN + t] = val;
    }
}

extern "C" void kernel_launch(void* const* d_in, const int* in_sizes, int n_in,
                              void* d_out, int out_size, void* d_ws, size_t ws_size,
                              hipStream_t stream) {
    (void)in_sizes; (void)n_in; (void)out_size; (void)d_ws; (void)ws_size;
    const float* mov = (const float*)d_in[0];
    const float* fix = (const float*)d_in[1];
    float* out = (float*)d_out;
    dim3 grid(VN / BT, VN / BY, VN / BX);   // (6, 24, 24)
    corr3d_wmma_kernel<<<grid, NTHREADS, 0, stream>>>(mov, fix, out);
}